// TTEmbedding_24464133718805
// MI455X (gfx1250) — compile-verified
//
#include <hip/hip_runtime.h>

typedef __attribute__((ext_vector_type(2))) float v2f;
typedef __attribute__((ext_vector_type(8))) float v8f;

#define NTOK   16384       // B*T = 4*4096
#define E_DIM  1024
#define NPQ    1280        // 40*32
#define M_FLOATS ((size_t)NPQ * 1024)   // swizzled M_pq table in workspace

// ---------------------------------------------------------------------------
// Stage 1: M_pq[x*8+y, c] = sum_b core0[0,x,p,b] * core1[b,y,q,c]
// Stored pre-swizzled in the WMMA 16x4 f32 A-operand layout:
//   ws[pq*1024 + (op*32 + lane)*2 + j]
//     op = tm*4 + tk   (tm: M-tile 0..3, tk: K-step 0..3)
//     row = lane%16 + 16*tm     (row = x*8 + y)
//     c   = 4*tk + j + 2*(lane/16)
// ---------------------------------------------------------------------------
__global__ __launch_bounds__(32)
void tt_stage1(const float* __restrict__ core0,   // (1,8,40,16)
               const float* __restrict__ core1,   // (16,8,32,16)
               float* __restrict__ M) {
  const int pq   = blockIdx.x;       // 0..1279
  const int p    = pq >> 5;          // /32
  const int q    = pq & 31;
  const int lane = threadIdx.x;      // one wave per (p,q)
  const int half = lane >> 4;
  const int rlo  = lane & 15;
  float* outp = M + (size_t)pq * 1024;
#pragma unroll
  for (int op = 0; op < 16; ++op) {
    const int tm = op >> 2, tk = op & 3;
    const int row = rlo + 16 * tm;   // = x*8 + y
    const int x = row >> 3, y = row & 7;
#pragma unroll
    for (int j = 0; j < 2; ++j) {
      const int c = 4 * tk + j + 2 * half;
      float acc = 0.f;
#pragma unroll
      for (int b = 0; b < 16; ++b)
        acc += core0[(x * 40 + p) * 16 + b] * core1[((b * 8 + y) * 32 + q) * 16 + c];
      outp[(op * 32 + lane) * 2 + j] = acc;
    }
  }
}

// ---------------------------------------------------------------------------
// Stage 2: per token, E(64x16) = M_pq(64x16) x C_r(16x16) via 16x
// V_WMMA_F32_16X16X4_F32 (4 M-tiles x 4 K-steps), one wave per token-iter.
// core2 staged in LDS pre-swizzled into the 4x16 f32 B-operand layout:
//   ldsC[ ((r*4 + kk)*32 + lane)*2 + j ],  k = 4*kk + j + 2*(lane/16),
//   n = lane%16,  value = core2[k_c*400 + n_z*25 + r]
// ---------------------------------------------------------------------------
__global__ __launch_bounds__(256)
void tt_stage2(const float* __restrict__ core2,   // (16,16,25,1)
               const int*   __restrict__ ids,     // (B,T) flat
               const float* __restrict__ M,       // swizzled table from stage 1
               float* __restrict__ out) {         // (B,T,E) flat
  __shared__ float ldsC[25 * 4 * 32 * 2];         // 6400 floats = 25.6 KB
  for (int f = threadIdx.x; f < 6400; f += 256) {
    const int j    = f & 1;
    const int lane = (f >> 1) & 31;
    const int kk   = (f >> 6) & 3;
    const int r    = f >> 8;
    const int c    = 4 * kk + j + 2 * (lane >> 4);
    const int z    = lane & 15;
    ldsC[f] = core2[c * 400 + z * 25 + r];
  }
  __syncthreads();

  const int lane = threadIdx.x & 31;
  const int wave = threadIdx.x >> 5;
  const int gw   = gridDim.x * 8;                 // total waves
  const int half = lane >> 4;
  const int nlo  = lane & 15;

  for (int t = blockIdx.x * 8 + wave; t < NTOK; t += gw) {   // wave-uniform
    const int v  = ids[t];
    const int pq = v / 25;
    const int r  = v - pq * 25;

    // A operands: 16 coalesced b64 loads from L2-resident table
    const v2f* Ab = (const v2f*)(M + (size_t)pq * 1024) + lane;
    v2f a[16];
#pragma unroll
    for (int op = 0; op < 16; ++op) a[op] = Ab[op * 32];

    // B operands: 4 b64 LDS loads
    const v2f* Bb = (const v2f*)ldsC + r * 128 + lane;
    v2f bm[4];
#pragma unroll
    for (int kk = 0; kk < 4; ++kk) bm[kk] = Bb[kk * 32];

    // D element (tile tm, m = j + 8*half, n = lane%16):
    //   e = (16*tm + j + 8*half)*16 + n  ->  tm*256 + j*16 + half*128 + n
    float* obase = out + (size_t)t * E_DIM + nlo + (half << 7);
#pragma unroll
    for (int tm = 0; tm < 4; ++tm) {
      v8f acc = {};
#pragma unroll
      for (int kk = 0; kk < 4; ++kk) {
        // 8 args: (neg_a, A, neg_b, B, c_mod, C, reuse_a, reuse_b)
        acc = __builtin_amdgcn_wmma_f32_16x16x4_f32(
            false, a[tm * 4 + kk], false, bm[kk], (short)0, acc, false, false);
      }
#pragma unroll
      for (int j = 0; j < 8; ++j) obase[tm * 256 + j * 16] = acc[j];
    }
  }
}

// ---------------------------------------------------------------------------
// Fallback (only if workspace is too small): direct triple contraction.
// ---------------------------------------------------------------------------
__global__ __launch_bounds__(256)
void tt_naive(const float* __restrict__ c0, const float* __restrict__ c1,
              const float* __restrict__ c2, const int* __restrict__ ids,
              float* __restrict__ out) {
  const int idx = blockIdx.x * blockDim.x + threadIdx.x;
  if (idx >= NTOK * E_DIM) return;
  const int t = idx >> 10, e = idx & 1023;
  const int x = e >> 7, y = (e >> 4) & 7, z = e & 15;
  const int v = ids[t];
  const int p = v / 800, rem = v - p * 800;
  const int q = rem / 25, r = rem - q * 25;
  float acc = 0.f;
  for (int b = 0; b < 16; ++b) {
    float inner = 0.f;
#pragma unroll
    for (int c = 0; c < 16; ++c)
      inner += c1[((b * 8 + y) * 32 + q) * 16 + c] * c2[c * 400 + z * 25 + r];
    acc += c0[(x * 40 + p) * 16 + b] * inner;
  }
  out[idx] = acc;
}

extern "C" void kernel_launch(void* const* d_in, const int* in_sizes, int n_in,
                              void* d_out, int out_size, void* d_ws, size_t ws_size,
                              hipStream_t stream) {
  const float* core0 = (const float*)d_in[0];   // 5120 f32
  const float* core1 = (const float*)d_in[1];   // 65536 f32
  const float* core2 = (const float*)d_in[2];   // 6400 f32
  const int*   ids   = (const int*)d_in[3];     // 16384 int32
  float* out = (float*)d_out;                   // 16384*1024 f32

  const size_t need = M_FLOATS * sizeof(float); // 5.25 MB
  if (ws_size >= need) {
    float* M = (float*)d_ws;
    tt_stage1<<<NPQ, 32, 0, stream>>>(core0, core1, M);
    tt_stage2<<<256, 256, 0, stream>>>(core2, ids, M, out);
  } else {
    tt_naive<<<(NTOK * E_DIM + 255) / 256, 256, 0, stream>>>(core0, core1, core2, ids, out);
  }
}